// Dempster_layer_3040836845749
// MI455X (gfx1250) — compile-verified
//
#include <hip/hip_runtime.h>
#include <cstdint>

// Problem constants (B=262144, P=64, K=11, f32).
#define PDIM   64
#define KDIM   11
#define TPB    256                       // one batch element per thread
#define TCH    2                         // p-rows per pipeline chunk
#define NCH    (PDIM / TCH)              // 32 chunks
#define CELEM  (TCH * KDIM)              // 22 floats per thread per chunk
#define LSTRIDE (CELEM + 1)              // 23 (odd -> conflict-free LDS banks)
#define BUF_FLOATS (TPB * LSTRIDE)       // 5888 floats per buffer
#define LDS_BYTES  (2 * BUF_FLOATS * 4)  // 47104 B dynamic LDS (double buffer)

#define ASYNC_WAIT_22() asm volatile("s_wait_asynccnt 22" ::: "memory")
#define ASYNC_WAIT_0()  asm volatile("s_wait_asynccnt 0"  ::: "memory")

// Issue one chunk: 22 cooperative B32 async global->LDS copies per thread.
// Flat element e = tid + 256*i -> row r = e/22 (batch), col j = e%22, so a
// wave's 32 lanes touch mostly-consecutive global addresses (coalesced),
// while the LDS destination uses the odd-stride swizzle. The r/j-derived
// base offsets are tid-only -> hoisted to registers by the compiler; only
// gchunk/bufo vary per call.
__device__ __forceinline__ void issue_chunk(const float* __restrict__ gbase,
                                            unsigned lds0, int tid, int c) {
  const unsigned bufo   = (unsigned)((c & 1) * (BUF_FLOATS * 4));
  const unsigned gchunk = (unsigned)(c * CELEM * 4);
#pragma unroll
  for (int i = 0; i < CELEM; ++i) {
    const int e = tid + i * TPB;
    const int r = e / CELEM;                 // batch row within block tile
    const int j = e - r * CELEM;             // float index within chunk
    const unsigned vgoff = (unsigned)(r * (PDIM * KDIM * 4)) + gchunk
                         + (unsigned)(j * 4);
    const unsigned vlds  = lds0 + bufo + (unsigned)((r * LSTRIDE + j) * 4);
    // GVS mode: mem_addr = SGPR64(gbase) + VGPR32(vgoff); LDS dest = VGPR(vlds)
    asm volatile("global_load_async_to_lds_b32 %0, %1, %2"
                 :: "v"(vlds), "v"(vgoff), "s"(gbase) : "memory");
  }
}

// One Dempster combine step: m <- normalize(m*m2 + m*w2 + w1*m2).
__device__ __forceinline__ void combine(float m[KDIM], const float* __restrict__ row) {
  float m2[KDIM];
#pragma unroll
  for (int k = 0; k < KDIM; ++k) m2[k] = row[k];
  const float w1 = m[KDIM - 1];
  const float w2 = m2[KDIM - 1];
  float cv[KDIM];
  float s = 0.0f;
#pragma unroll
  for (int k = 0; k < KDIM; ++k) {
    float v = m[k] * m2[k] + m[k] * w2 + w1 * m2[k];   // 2 fma + 1 mul
    cv[k] = v;
    s += v;
  }
  const float inv = 1.0f / s;                          // one accurate divide
#pragma unroll
  for (int k = 0; k < KDIM; ++k) m[k] = cv[k] * inv;
}

__global__ __launch_bounds__(TPB)
void dempster_scan_kernel(const float* __restrict__ in,
                          float* __restrict__ out) {
  extern __shared__ float smem[];
  const int tid = threadIdx.x;
  const int b0  = blockIdx.x * TPB;

  // Flat LDS addresses carry the LDS byte offset in the low 32 bits; dynamic
  // LDS starts at 0 here (no static shared in use).
  const unsigned lds0 = (unsigned)(uintptr_t)(void*)smem;
  const float* gbase = in + (size_t)b0 * (PDIM * KDIM);

  float m[KDIM];

  // Prime pipeline: chunks 0 and 1 in flight (44 outstanding async ops/wave).
  issue_chunk(gbase, lds0, tid, 0);
  issue_chunk(gbase, lds0, tid, 1);

  // ---- peeled chunk 0: initialize m from p=0, combine p=1 ----
  ASYNC_WAIT_22();          // oldest 22 (chunk 0) landed; chunk 1 may stream
  __syncthreads();          // cross-thread copy visibility
  {
    const float* buf = smem + tid * LSTRIDE;
#pragma unroll
    for (int k = 0; k < KDIM; ++k) m[k] = buf[k];
    combine(m, buf + KDIM);
  }
  __syncthreads();          // all reads of buffer 0 done
  issue_chunk(gbase, lds0, tid, 2);

  // ---- steady state: chunks 1 .. NCH-2, branch-free wait, unroll x2 so the
  // (c&1) buffer select and bufo fold to compile-time constants ----
#pragma unroll 2
  for (int c = 1; c < NCH - 1; ++c) {
    ASYNC_WAIT_22();        // drain oldest in-flight chunk (in-order retire)
    __syncthreads();
    const float* buf = smem + (c & 1) * BUF_FLOATS + tid * LSTRIDE;
    combine(m, buf);
    combine(m, buf + KDIM);
    __syncthreads();        // buffer reusable
    if (c + 2 < NCH) issue_chunk(gbase, lds0, tid, c + 2);
  }

  // ---- peeled last chunk ----
  ASYNC_WAIT_0();
  __syncthreads();
  {
    const float* buf = smem + ((NCH - 1) & 1) * BUF_FLOATS + tid * LSTRIDE;
    combine(m, buf);
    combine(m, buf + KDIM);
  }

  // Stage results through LDS (stride 12) for coalesced global stores.
  // Staging region (3072 floats) lives in buffer 0; the last chunk was read
  // from buffer 1, so no barrier is needed before writing.
#pragma unroll
  for (int k = 0; k < KDIM; ++k) smem[tid * (KDIM + 1) + k] = m[k];
  __syncthreads();

  float* gout = out + (size_t)b0 * KDIM;
#pragma unroll
  for (int i = 0; i < KDIM; ++i) {
    const int e = tid + i * TPB;          // e = r*11 + j -> out[b0*11 + e]
    const int r = e / KDIM;
    const int j = e - r * KDIM;
    gout[e] = smem[r * (KDIM + 1) + j];
  }
}

extern "C" void kernel_launch(void* const* d_in, const int* in_sizes, int n_in,
                              void* d_out, int out_size, void* d_ws, size_t ws_size,
                              hipStream_t stream) {
  (void)n_in; (void)out_size; (void)d_ws; (void)ws_size;
  const float* in = (const float*)d_in[0];
  float* out = (float*)d_out;
  const int nbatch = in_sizes[0] / (PDIM * KDIM);   // 262144
  const int blocks = nbatch / TPB;                  // 1024
  dempster_scan_kernel<<<blocks, TPB, LDS_BYTES, stream>>>(in, out);
}